// DigitCaps_627065225406
// MI455X (gfx1250) — compile-verified
//
#include <hip/hip_runtime.h>
#include <math.h>

// DigitCaps dynamic routing, fused (u_hat never materialized).
// BS=64, R=2048, J=32, I=16, O=32, 3 routing iterations.
// Core GEMM per r: [64 x 16] @ [16 x 1024] via V_WMMA_F32_16X16X4_F32.

typedef __attribute__((ext_vector_type(2))) float v2f;
typedef __attribute__((ext_vector_type(8))) float v8f;

#define BSZ   64
#define RTOT  2048
#define JCAP  32
#define ICAP  16
#define OCAP  32
#define JO    1024   // J*O

__device__ __forceinline__ float wave_reduce_add(float x) {
#pragma unroll
  for (int m = 16; m >= 1; m >>= 1) x += __shfl_xor(x, m, 32);
  return x;
}
__device__ __forceinline__ float wave_reduce_max(float x) {
#pragma unroll
  for (int m = 16; m >= 1; m >>= 1) x = fmaxf(x, __shfl_xor(x, m, 32));
  return x;
}

// One routing pass:
//   if (!first_iter): b_new[r,j] = b_in[r,j] + (1/BS) * sum_{b,o} u_hat[b,r,j,o]*v_prev[b,j,o]
//                     c[r,j] = softmax_j(b_new[r,:]);  b_out = b_new
//   else:             c = 1/J
//   s_part[chunk][b,j,o] = sum_{r in chunk} c[r,j] * u_hat[b,r,j,o]
__launch_bounds__(256)
__global__ void routing_pass(const float* __restrict__ in_caps,  // [64,2048,16]
                             const float* __restrict__ Wt,       // [2048,32,32,16]
                             const float* __restrict__ b_in,     // [2048,32] or nullptr (== zeros)
                             const float* __restrict__ v_prev,   // [64,32,32] (unused if first_iter)
                             float* __restrict__ b_out,          // [2048,32]
                             float* __restrict__ s_part,         // [nchunk][64*1024]
                             int rc, int first_iter) {
  __shared__ float dot_lds[64][33];  // rc <= 64
  __shared__ float c_lds[64][33];

  const int chunk = blockIdx.x;
  const int r0    = chunk * rc;
  const int lane  = threadIdx.x & 31;
  const int wave  = threadIdx.x >> 5;
  const int half  = lane >> 4;   // 0: lanes 0-15, 1: lanes 16-31
  const int l16   = lane & 15;
  const int jbase = wave * 4;    // wave exclusively owns j in [jbase, jbase+4)

  if (!first_iter) {
    // ---------- Phase 1: logit updates via WMMA + per-lane dot with v_prev ----------
    for (int rr = 0; rr < rc; ++rr) {
      const int r = r0 + rr;
      const float* inr = in_caps + (size_t)r * ICAP;
      const float* wr  = Wt + (size_t)r * (JO * ICAP);
      if (rr + 1 < rc) {  // warm next r's W slab (64KB) -> global_prefetch_b8
        __builtin_prefetch(wr + JO * ICAP + (size_t)threadIdx.x * 64, 0, 0);
        __builtin_prefetch(wr + JO * ICAP + 8192 + (size_t)threadIdx.x * 64, 0, 0);
      }
      float bpart[4] = {0.f, 0.f, 0.f, 0.f};
#pragma unroll
      for (int bt = 0; bt < 4; ++bt) {
        const int b0 = bt * 16;
        v2f a4[4];
#pragma unroll
        for (int ks = 0; ks < 4; ++ks) {  // A frag: lane holds M=l16, K = 4ks + 2*half {,+1}
          const int k0 = 4 * ks + 2 * half;
          a4[ks] = *(const v2f*)(inr + (size_t)(b0 + l16) * (RTOT * ICAP) + k0);
        }
#pragma unroll
        for (int jl = 0; jl < 4; ++jl) {
          const int j = jbase + jl;
#pragma unroll
          for (int oh = 0; oh < 2; ++oh) {
            const int jo0 = j * OCAP + oh * 16;
            v8f acc = {0.f, 0.f, 0.f, 0.f, 0.f, 0.f, 0.f, 0.f};
#pragma unroll
            for (int ks = 0; ks < 4; ++ks) {  // B frag: lane holds N=l16, K = 4ks + 2*half {,+1}
              const int k0 = 4 * ks + 2 * half;
              v2f bf = *(const v2f*)(wr + (size_t)(jo0 + l16) * ICAP + k0);
              acc = __builtin_amdgcn_wmma_f32_16x16x4_f32(
                  false, a4[ks], false, bf, (short)0, acc, false, false);
            }
            // D element (m,n): m = v + 8*half, n = l16   ->  dot with v_prev[b,j,o]
            const float* vp = v_prev + (size_t)(b0 + half * 8) * JO + jo0 + l16;
            float d = 0.f;
#pragma unroll
            for (int m = 0; m < 8; ++m) d += acc[m] * vp[(size_t)m * JO];
            bpart[jl] += d;
          }
        }
      }
#pragma unroll
      for (int jl = 0; jl < 4; ++jl) {
        float s = wave_reduce_add(bpart[jl]);
        if (lane == 0) dot_lds[rr][jbase + jl] = s;
      }
    }
    __syncthreads();
    // ---------- softmax over J (lane = j), rows distributed over waves ----------
    for (int rr = wave; rr < rc; rr += 8) {
      const int r = r0 + rr;
      float bv = dot_lds[rr][lane] * (1.0f / (float)BSZ);
      if (b_in) bv += b_in[(size_t)r * JCAP + lane];
      b_out[(size_t)r * JCAP + lane] = bv;
      float mx = wave_reduce_max(bv);
      float e  = __expf(bv - mx);
      float se = wave_reduce_add(e);
      c_lds[rr][lane] = e / se;
    }
    __syncthreads();
  }

  // ---------- Phase 2: s accumulation; fold c[r,j] into B so WMMA C chains over r ----------
  float* sp = s_part + (size_t)chunk * (BSZ * JO);
  for (int jl = 0; jl < 4; ++jl) {
    const int j = jbase + jl;
    for (int oh = 0; oh < 2; ++oh) {
      const int jo0 = j * OCAP + oh * 16;
      for (int bt = 0; bt < 4; ++bt) {
        const int b0 = bt * 16;
        v8f acc = {0.f, 0.f, 0.f, 0.f, 0.f, 0.f, 0.f, 0.f};
        for (int rr = 0; rr < rc; ++rr) {
          const int r = r0 + rr;
          const float cs = first_iter ? (1.0f / (float)JCAP) : c_lds[rr][j];
          const float* inr = in_caps + (size_t)r * ICAP;
          const float* wr  = Wt + (size_t)r * (JO * ICAP);
#pragma unroll
          for (int ks = 0; ks < 4; ++ks) {
            const int k0 = 4 * ks + 2 * half;
            v2f a  = *(const v2f*)(inr + (size_t)(b0 + l16) * (RTOT * ICAP) + k0);
            v2f bf = *(const v2f*)(wr + (size_t)(jo0 + l16) * ICAP + k0);
            bf.x *= cs;
            bf.y *= cs;
            acc = __builtin_amdgcn_wmma_f32_16x16x4_f32(
                false, a, false, bf, (short)0, acc, false, false);
          }
        }
        float* sb = sp + (size_t)(b0 + half * 8) * JO + jo0 + l16;
#pragma unroll
        for (int m = 0; m < 8; ++m) sb[(size_t)m * JO] = acc[m];
      }
    }
  }
}

// Fold chunk partials and apply squash along O. One wave per (b,j) row, lane = o.
__launch_bounds__(256)
__global__ void reduce_squash(const float* __restrict__ s_part, int nchunk,
                              float* __restrict__ v_out) {
  const int lane = threadIdx.x & 31;
  const int wave = threadIdx.x >> 5;
  const int row  = blockIdx.x * 8 + wave;        // row = b*32 + j, in [0, 2048)
  const size_t base = (size_t)row * OCAP + lane;
  float s = 0.f;
  for (int c = 0; c < nchunk; ++c) s += s_part[(size_t)c * (BSZ * JO) + base];
  float sq = wave_reduce_add(s * s);
  float vv = (sq / (1.0f + sq)) * (s / sqrtf(sq));
  v_out[base] = vv;
}

extern "C" void kernel_launch(void* const* d_in, const int* in_sizes, int n_in,
                              void* d_out, int out_size, void* d_ws, size_t ws_size,
                              hipStream_t stream) {
  const float* in_caps = (const float*)d_in[0];  // [64,2048,16]
  const float* Wt      = (const float*)d_in[1];  // [2048,32,32,16]
  float* out = (float*)d_out;                    // [64,32,32,1] flat = 65536 f32

  // Workspace: s_part[nchunk][64*1024] + v_buf[64*1024] + b1[2048*32] + b2[2048*32]
  const size_t fixed_f = (size_t)(BSZ * JO) + 2u * (RTOT * JCAP);
  int nchunk = 128;
  while (nchunk > 32 &&
         (fixed_f + (size_t)nchunk * (BSZ * JO)) * sizeof(float) > ws_size)
    nchunk >>= 1;
  const int rc = RTOT / nchunk;  // 16 / 32 / 64

  float* s_part = (float*)d_ws;
  float* v_buf  = s_part + (size_t)nchunk * (BSZ * JO);
  float* b1     = v_buf + (BSZ * JO);
  float* b2     = b1 + (RTOT * JCAP);

  dim3 blk(256), grid(nchunk), rgrid((BSZ * JCAP) / 8);

  // it 0: c = 1/J; v0
  routing_pass<<<grid, blk, 0, stream>>>(in_caps, Wt, nullptr, nullptr, b1, s_part, rc, 1);
  reduce_squash<<<rgrid, blk, 0, stream>>>(s_part, nchunk, v_buf);
  // it 1: b1 = 0 + dot(u_hat, v0)/BS; c = softmax(b1); v1
  routing_pass<<<grid, blk, 0, stream>>>(in_caps, Wt, nullptr, v_buf, b1, s_part, rc, 0);
  reduce_squash<<<rgrid, blk, 0, stream>>>(s_part, nchunk, v_buf);
  // it 2: b2 = b1 + dot(u_hat, v1)/BS; c = softmax(b2); v2 -> out
  routing_pass<<<grid, blk, 0, stream>>>(in_caps, Wt, b1, v_buf, b2, s_part, rc, 0);
  reduce_squash<<<rgrid, blk, 0, stream>>>(s_part, nchunk, out);
}